// MixVar_32083405701670
// MI455X (gfx1250) — compile-verified
//
#include <hip/hip_runtime.h>
#include <stddef.h>

// ---------------------------------------------------------------------------
// MixVar lookup for MI455X (gfx1250, wave32).
// Pure-bandwidth gather: ~259 MB moved -> ~11 us at 23.3 TB/s HBM.
// k_gather is defined FIRST so the disasm snippet shows the hot kernel
// (verify global_load_b128 + global_store_b128 th:TH_STORE_NT).
// ---------------------------------------------------------------------------

typedef __attribute__((ext_vector_type(4))) float v4f;
typedef __attribute__((ext_vector_type(4))) int   v4i;

#define TILE 2048      // rows per scan tile
#define DVEC 32        // 128 floats per row = 32 float4
#define RPW  8         // output rows per wave

// ---------------- the bandwidth-critical gather (pass 4) ----------------
// One wave handles RPW output rows; 32 lanes x float4 = one full 512B row
// per b128 access. Lanes 0..7 fetch index/info once and broadcast with
// wave32 shuffles. Gather loads stay temporal (192MB L2 holds ~half the
// 384MB of tables + all 2MB of info); the 128MB output stream uses
// non-temporal stores so it does not evict the tables.

__global__ void __launch_bounds__(256)
k_gather(const v4f* __restrict__ X, const v4f* __restrict__ W,
         const int* __restrict__ index, const int* __restrict__ info,
         v4f* __restrict__ out, int mtotal) {
  const int tid  = blockIdx.x * blockDim.x + threadIdx.x;
  const int wave = tid >> 5;
  const int lane = tid & 31;
  const int base = wave * RPW;

  // Stream-ahead prefetch of the index array (gfx1250 global_prefetch_b8).
  if (lane == 0 && base + 4096 < mtotal) {
    __builtin_prefetch(&index[base + 4096], 0, 1);
  }

  int idxr = 0, pr = 0;
  if (lane < RPW) {
    const int m = base + lane;
    if (m < mtotal) {
      idxr = __builtin_nontemporal_load(index + m);  // index streamed once
      pr   = info[idxr];                             // info: temporal, L2-hot
    }
  }

  if (base + RPW <= mtotal) {
    // Fast path: no per-row bounds checks; 8 independent gather+store pairs.
#pragma unroll
    for (int r = 0; r < RPW; ++r) {
      const int idx = __shfl(idxr, r, 32);
      const int p   = __shfl(pr,   r, 32);
      const v4f* src = (p & 1) ? (X + (size_t)idx * DVEC)
                               : (W + (size_t)(p >> 1) * DVEC);
      const v4f v = src[lane];                             // global_load_b128
      __builtin_nontemporal_store(v, out + (size_t)(base + r) * DVEC + lane);
    }
  } else {
#pragma unroll
    for (int r = 0; r < RPW; ++r) {
      const int m = base + r;
      if (m >= mtotal) break;
      const int idx = __shfl(idxr, r, 32);
      const int p   = __shfl(pr,   r, 32);
      const v4f* src = (p & 1) ? (X + (size_t)idx * DVEC)
                               : (W + (size_t)(p >> 1) * DVEC);
      const v4f v = src[lane];
      __builtin_nontemporal_store(v, out + (size_t)m * DVEC + lane);
    }
  }
}

// ---------------- wave32 / block scan helpers ----------------

__device__ __forceinline__ int wave_incl_scan(int v, int lane) {
#pragma unroll
  for (int d = 1; d < 32; d <<= 1) {
    int t = __shfl_up(v, d, 32);
    if (lane >= d) v += t;
  }
  return v;
}

// Exclusive scan over all threads of a (<=256-thread) block, thread order.
__device__ __forceinline__ int block_excl_scan(int v, int* wsum) {
  const int lane = threadIdx.x & 31;
  const int wid  = threadIdx.x >> 5;
  int inc = wave_incl_scan(v, lane);
  if (lane == 31) wsum[wid] = inc;
  __syncthreads();
  if (wid == 0) {
    const int nw = blockDim.x >> 5;
    int s = (lane < nw) ? wsum[lane] : 0;
    s = wave_incl_scan(s, lane);
    if (lane < nw) wsum[lane] = s;
  }
  __syncthreads();
  const int off = (wid > 0) ? wsum[wid - 1] : 0;
  return off + inc - v;  // exclusive prefix for this thread
}

// ---------------- pass 1: per-tile counts of non-const rows ----------------
// Each thread owns 8 contiguous ints (32B, b128-aligned): full tiles use two
// unguarded vector loads so no per-element exec-mask branches are emitted.

__global__ void __launch_bounds__(256)
k_partials(const int* __restrict__ mask, int* __restrict__ partials, int n) {
  __shared__ int wsum[8];
  const int tbase = blockIdx.x * TILE + threadIdx.x * 8;
  int s = 0;
  if (tbase + 8 <= n) {
    const v4i a = *(const v4i*)(mask + tbase);
    const v4i b = *(const v4i*)(mask + tbase + 4);
#pragma unroll
    for (int k = 0; k < 4; ++k) s += (a[k] == 0) + (b[k] == 0);
  } else {
#pragma unroll
    for (int k = 0; k < 8; ++k) {
      const int i = tbase + k;
      if (i < n) s += (mask[i] == 0);
    }
  }
  const int ex = block_excl_scan(s, wsum);
  if (threadIdx.x == blockDim.x - 1) partials[blockIdx.x] = ex + s;
}

// ---------------- pass 2: exclusive scan of tile partials ----------------

__global__ void __launch_bounds__(256)
k_scan(const int* __restrict__ partials, int* __restrict__ offsets, int ntiles) {
  __shared__ int wsum[8];
  const int v  = (threadIdx.x < ntiles) ? partials[threadIdx.x] : 0;
  const int ex = block_excl_scan(v, wsum);
  if (threadIdx.x < ntiles) offsets[threadIdx.x] = ex;
}

// ---------------- pass 3: packed per-row descriptor ----------------
// info[i] = 1                       if row i is constant  (low bit set)
//         = clip(var_pos[i]) << 1   otherwise             (low bit clear)

__global__ void __launch_bounds__(256)
k_info(const int* __restrict__ mask, const int* __restrict__ offsets,
       int* __restrict__ info, int n, int nvar) {
  __shared__ int wsum[8];
  const int tbase = blockIdx.x * TILE + threadIdx.x * 8;
  int m[8];
  int tot = 0;
  if (tbase + 8 <= n) {
    const v4i a = *(const v4i*)(mask + tbase);
    const v4i b = *(const v4i*)(mask + tbase + 4);
#pragma unroll
    for (int k = 0; k < 4; ++k) { m[k] = a[k]; m[k + 4] = b[k]; }
  } else {
#pragma unroll
    for (int k = 0; k < 8; ++k) {
      const int i = tbase + k;
      m[k] = (i < n) ? mask[i] : 1;
    }
  }
#pragma unroll
  for (int k = 0; k < 8; ++k) tot += (m[k] == 0);

  int run = offsets[blockIdx.x] + block_excl_scan(tot, wsum);

  if (tbase + 8 <= n) {
    v4i oa, ob;
#pragma unroll
    for (int k = 0; k < 8; ++k) {
      run += (m[k] == 0);
      int vp = run - 1;
      vp = vp < 0 ? 0 : vp;
      vp = vp >= nvar ? nvar - 1 : vp;
      const int v = m[k] ? 1 : (vp << 1);
      if (k < 4) oa[k] = v; else ob[k - 4] = v;
    }
    *(v4i*)(info + tbase)     = oa;   // b128 stores, fully coalesced per 8-chunk
    *(v4i*)(info + tbase + 4) = ob;
  } else {
#pragma unroll
    for (int k = 0; k < 8; ++k) {
      const int i = tbase + k;
      if (i < n) {
        run += (m[k] == 0);
        int vp = run - 1;
        vp = vp < 0 ? 0 : vp;
        vp = vp >= nvar ? nvar - 1 : vp;
        info[i] = m[k] ? 1 : (vp << 1);
      }
    }
  }
}

// ---------------- host launcher ----------------

extern "C" void kernel_launch(void* const* d_in, const int* in_sizes, int n_in,
                              void* d_out, int out_size, void* d_ws, size_t ws_size,
                              hipStream_t stream) {
  const float* X     = (const float*)d_in[0];   // [N, 128] fp32
  const int*   mask  = (const int*)  d_in[1];   // [N] int32
  const float* W     = (const float*)d_in[2];   // [N/2, 128] fp32
  const int*   index = (const int*)  d_in[3];   // [M] int32
  float*       out   = (float*)d_out;           // [M, 128] fp32

  const int N    = in_sizes[1];
  const int M    = in_sizes[3];
  const int nvar = in_sizes[2] / 128;

  // Workspace layout: info[N] | partials[ntiles] | offsets[ntiles]
  int* info        = (int*)d_ws;
  const int ntiles = (N + TILE - 1) / TILE;     // 256 for N = 524288
  int* partials    = info + N;
  int* offsets     = partials + ntiles;

  hipLaunchKernelGGL(k_partials, dim3(ntiles), dim3(256), 0, stream,
                     mask, partials, N);
  hipLaunchKernelGGL(k_scan, dim3(1), dim3(256), 0, stream,
                     partials, offsets, ntiles);
  hipLaunchKernelGGL(k_info, dim3(ntiles), dim3(256), 0, stream,
                     mask, offsets, info, N, nvar);

  const int waves  = (M + RPW - 1) / RPW;
  const int blocks = (waves * 32 + 255) / 256;
  hipLaunchKernelGGL(k_gather, dim3(blocks), dim3(256), 0, stream,
                     (const v4f*)X, (const v4f*)W, index, info,
                     (v4f*)out, M);
}